// SpatialGCN_19507741458586
// MI455X (gfx1250) — compile-verified
//
#include <hip/hip_runtime.h>

typedef float v2f __attribute__((ext_vector_type(2)));
typedef float v8f __attribute__((ext_vector_type(8)));
typedef int   vi4 __attribute__((vector_size(16)));   // matches builtin param type

#define B_ 16
#define T_ 48
#define N_ 512
#define F_ 64
#define H_ 128
#define E_ 8192
#define EPN (E_ + N_)        // 8704 edges incl. self loops
#define XS_STRIDE 68         // padded row stride (floats) -> conflict-free LDS gathers
#define WT_STRIDE 68         // padded stride for transposed W
#define NWAVES 8
#define TILES (N_ / 16)      // 32 row-tiles of 16 dst nodes

#ifndef __has_builtin
#define __has_builtin(x) 0
#endif

#if __has_builtin(__builtin_amdgcn_global_load_async_to_lds_b128)
#define HAVE_ASYNC_LDS 1
#else
#define HAVE_ASYNC_LDS 0
#endif

#define AS1 __attribute__((address_space(1)))
#define AS3 __attribute__((address_space(3)))

// ---------------------------------------------------------------------------
// Prep: build deterministic CSR-by-dst (rowptr, col, val=dinv[src]*dinv[dst])
// No atomics: per-node scans over an LDS copy of dst[] give a stable ordering
// (edges in index order, self-loop appended last, matching the reference).
// ---------------------------------------------------------------------------
__global__ void __launch_bounds__(N_) gcn_prep(const long long* __restrict__ ei,
                                               int* __restrict__ rowptr,
                                               int* __restrict__ col,
                                               float* __restrict__ val) {
    __shared__ int   dsts[E_];    // 32 KB
    __shared__ float dinv[N_];
    __shared__ int   degs[N_];
    __shared__ int   rp[N_ + 1];

    const int t = threadIdx.x;    // 512 threads, t == node id
    for (int e = t; e < E_; e += N_) dsts[e] = (int)ei[E_ + e];
    __syncthreads();

    int cnt = 1;                  // self loop
    for (int e = 0; e < E_; ++e) cnt += (dsts[e] == t) ? 1 : 0;
    degs[t] = cnt;
    dinv[t] = rsqrtf((float)cnt);
    __syncthreads();

    if (t == 0) {
        int a = 0;
        for (int i = 0; i < N_; ++i) { rp[i] = a; a += degs[i]; }
        rp[N_] = a;
    }
    __syncthreads();

    rowptr[t] = rp[t];
    if (t == 0) rowptr[N_] = rp[N_];

    int w = rp[t];
    const float dt = dinv[t];
    for (int e = 0; e < E_; ++e) {
        if (dsts[e] == t) {
            int s = (int)ei[e];
            col[w] = s;
            val[w] = dinv[s] * dt;
            ++w;
        }
    }
    col[w] = t;                   // self loop last (reference concat order)
    val[w] = dt * dt;
}

// ---------------------------------------------------------------------------
// Main: one block per (b,t) graph. Async-staged X slab in LDS, CSR gather-
// aggregate, then fp32 WMMA (16x16x4) GEMM tile (agg[16x64] x W[64x128]) + b.
// ---------------------------------------------------------------------------
__global__ void __launch_bounds__(256) gcn_main(const float* __restrict__ x,
                                                const float* __restrict__ W,
                                                const float* __restrict__ bias,
                                                const int* __restrict__ rowptr,
                                                const int* __restrict__ col,
                                                const float* __restrict__ val,
                                                float* __restrict__ out) {
    extern __shared__ float smem[];
    float* Wt = smem;                        // 128*68 = 8704 floats (W transposed)
    float* xs = Wt + H_ * WT_STRIDE;         // 512*68 = 34816 floats
    float* sc = xs + N_ * XS_STRIDE;         // 8*16*68 = 8704 floats

    const int bt   = blockIdx.x;
    const int tid  = threadIdx.x;
    const int lane = tid & 31;
    const int wave = tid >> 5;
    const int l15  = lane & 15;

    const float* xg = x   + (size_t)bt * (N_ * F_);
    float*       og = out + (size_t)bt * (N_ * H_);

    // ---- Stage X slab into padded LDS rows (dominant HBM traffic) ----
#if HAVE_ASYNC_LDS
    for (int i = tid; i < (N_ * F_) / 4; i += 256) {
        const int row = i >> 4;              // 16 float4 per row
        const int c4  = i & 15;
        __builtin_amdgcn_global_load_async_to_lds_b128(
            (AS1 vi4*)(xg + i * 4),
            (AS3 vi4*)(xs + row * XS_STRIDE + c4 * 4), 0, 0);
    }
#else
    for (int i = tid; i < (N_ * F_) / 4; i += 256) {
        const int row = i >> 4;
        const int c4  = i & 15;
        float4 v = ((const float4*)xg)[i];
        *(float4*)(xs + row * XS_STRIDE + c4 * 4) = v;
    }
#endif

    // ---- Stage W transposed: Wt[h*68 + f] (enables b64 B-fragment loads) ----
    for (int i = tid; i < (F_ * H_) / 4; i += 256) {
        const int f  = i >> 5;               // 32 float4 per W row (H=128)
        const int h4 = i & 31;
        float4 v = ((const float4*)W)[i];
        Wt[(h4 * 4 + 0) * WT_STRIDE + f] = v.x;
        Wt[(h4 * 4 + 1) * WT_STRIDE + f] = v.y;
        Wt[(h4 * 4 + 2) * WT_STRIDE + f] = v.z;
        Wt[(h4 * 4 + 3) * WT_STRIDE + f] = v.w;
    }

    // ---- Preload bias per lane (constant across tiles; folded into C init) ----
    float bv[8];
#pragma unroll
    for (int nt = 0; nt < 8; ++nt) bv[nt] = bias[nt * 16 + l15];

#if HAVE_ASYNC_LDS
#if __has_builtin(__builtin_amdgcn_s_wait_asynccnt)
    __builtin_amdgcn_s_wait_asynccnt(0);
#else
    asm volatile("s_wait_asynccnt 0x0" ::: "memory");
#endif
#endif
    __syncthreads();

    float* mysc = sc + wave * (16 * XS_STRIDE);
    const int hlf  = lane & 1;               // which 32-col half this lane aggregates
    const int arow = lane >> 1;              // 0..15: dst row within tile
    const int koff = (lane < 16) ? 0 : 2;    // fp32 A/B fragment K sub-offset

    for (int tile = wave; tile < TILES; tile += NWAVES) {
        const int rowbase = tile * 16;

        // ---- CSR gather-aggregate: agg[arow, hlf*32 .. hlf*32+31] ----
        float acc[32];
#pragma unroll
        for (int j = 0; j < 32; ++j) acc[j] = 0.0f;
        const int rs = rowptr[rowbase + arow];
        const int re = rowptr[rowbase + arow + 1];
        for (int p = rs; p < re; ++p) {
            const int   s  = col[p];
            const float nv = val[p];
            const float* srcp = xs + s * XS_STRIDE + hlf * 32;
#pragma unroll
            for (int j4 = 0; j4 < 8; ++j4) {
                float4 v = *(const float4*)(srcp + j4 * 4);
                acc[j4 * 4 + 0] += nv * v.x;
                acc[j4 * 4 + 1] += nv * v.y;
                acc[j4 * 4 + 2] += nv * v.z;
                acc[j4 * 4 + 3] += nv * v.w;
            }
        }
        // spill aggregated tile to wave-private LDS scratch (re-layout for WMMA)
        float* dstp = mysc + arow * XS_STRIDE + hlf * 32;
#pragma unroll
        for (int j4 = 0; j4 < 8; ++j4)
            *(float4*)(dstp + j4 * 4) =
                make_float4(acc[j4 * 4 + 0], acc[j4 * 4 + 1],
                            acc[j4 * 4 + 2], acc[j4 * 4 + 3]);

        // ---- A fragments: 16x4 fp32 per K-step, lane<16 holds K=4k,4k+1 ----
        v2f a[16];
#pragma unroll
        for (int k = 0; k < 16; ++k)
            a[k] = *(const v2f*)(mysc + l15 * XS_STRIDE + 4 * k + koff);

        // ---- 8 N-tiles of 16 cols, 16 K-steps each ----
#pragma unroll
        for (int nt = 0; nt < 8; ++nt) {
            const float b0 = bv[nt];
            v8f c = {b0, b0, b0, b0, b0, b0, b0, b0};   // bias folded into C
#pragma unroll
            for (int k = 0; k < 16; ++k) {
                // B fragment: contiguous pair from transposed W -> single b64
                v2f bf = *(const v2f*)(Wt + (nt * 16 + l15) * WT_STRIDE + 4 * k + koff);
                c = __builtin_amdgcn_wmma_f32_16x16x4_f32(
                        false, a[k], false, bf, (short)0, c, false, false);
            }
            const int rofs = (lane < 16) ? 0 : 8;       // C/D layout: M=i / M=8+i
#pragma unroll
            for (int i = 0; i < 8; ++i) {
                const int row = rowbase + i + rofs;
                og[row * H_ + nt * 16 + l15] = c[i];
            }
        }
    }
}

// ---------------------------------------------------------------------------
extern "C" void kernel_launch(void* const* d_in, const int* in_sizes, int n_in,
                              void* d_out, int out_size, void* d_ws, size_t ws_size,
                              hipStream_t stream) {
    (void)in_sizes; (void)n_in; (void)out_size; (void)ws_size;

    const float*     x  = (const float*)d_in[0];
    const long long* ei = (const long long*)d_in[1];  // int64 edge_index [2,E]
    const float*     W  = (const float*)d_in[2];
    const float*     b  = (const float*)d_in[3];
    float*           out = (float*)d_out;

    // ws layout: rowptr[1024 ints] | col[EPN ints] | val[EPN floats]
    int*   rowptr = (int*)d_ws;
    int*   col    = rowptr + 1024;
    float* val    = (float*)(col + EPN);

    gcn_prep<<<1, N_, 0, stream>>>(ei, rowptr, col, val);

    const size_t shmem =
        (size_t)(H_ * WT_STRIDE + N_ * XS_STRIDE + NWAVES * 16 * XS_STRIDE) * sizeof(float);
    (void)hipFuncSetAttribute(reinterpret_cast<const void*>(gcn_main),
                              hipFuncAttributeMaxDynamicSharedMemorySize, (int)shmem);
    gcn_main<<<B_ * T_, 256, shmem, stream>>>(x, W, b, rowptr, col, val, out);
}